// TypedEventLog_369367187861
// MI455X (gfx1250) — compile-verified
//
#include <hip/hip_runtime.h>
#include <math.h>

// ---------------- problem constants ----------------
#define Bn 8
#define Tn 4096
#define Dn 1024
#define En 64
#define Kn 32
#define NTYPES 7
#define MAXTIME 512
#define BT (Bn * Tn)           // 32768
#define THRv 0.4f
#define NEGV (-1.0e30f)

#define WPITCH 34              // transposed LDS tile pitch (even -> 8B aligned b64 reads)

typedef __attribute__((ext_vector_type(2))) float v2f;
typedef __attribute__((ext_vector_type(8))) float v8f;

__device__ __forceinline__ v8f wmma_f32(v2f a, v2f b, v8f c) {
    // D = A(16x4) * B(4x16) + C(16x16), fp32, wave32
    return __builtin_amdgcn_wmma_f32_16x16x4_f32(false, a, false, b, (short)0, c, false, false);
}

__device__ __forceinline__ float gelu_exact(float x) {
    return 0.5f * x * (1.0f + erff(x * 0.70710678118654752f));
}

// async global->LDS copy of one dword (per-lane LDS dest => transpose during copy)
__device__ __forceinline__ void async_copy_b32(uint32_t lds_byte_off, const float* g) {
    asm volatile("global_load_async_to_lds_b32 %0, %1, off"
                 :: "v"(lds_byte_off), "v"(g)
                 : "memory");
}
__device__ __forceinline__ void wait_async0() {
    asm volatile("s_wait_asynccnt 0x0" ::: "memory");
}

// stage W[k0+kk][n0+cc]  ->  ldsT[cc*WPITCH + kk]   (2048 elems, 8 per thread)
// global side is coalesced (cc contiguous); LDS side is the transposed scatter.
__device__ __forceinline__ void stage_w_async(const float* __restrict__ W, int N,
                                              int k0, int n0,
                                              const float* ldsT, int tid)
{
#pragma unroll
    for (int r = 0; r < 8; ++r) {
        const int idx = r * 256 + tid;          // 0..2047
        const int cc = idx & 63, kk = idx >> 6;
        const float* g = W + (size_t)(k0 + kk) * N + (n0 + cc);
        const uint32_t loff = (uint32_t)(size_t)(ldsT + cc * WPITCH + kk);
        async_copy_b32(loff, g);
    }
}

// =====================================================================
// Kernel 1: fused  h = gelu(seq@W1 + b1);  etl = h@W2 + b2;  scores
// block = 256 threads (8 waves) owns 128 rows; h never touches HBM.
// W1 tile: async double-buffered, transposed in LDS.
// =====================================================================
__global__ __launch_bounds__(256) void fused_h_etl_kernel(
    const float* __restrict__ seq,
    const float* __restrict__ W1, const float* __restrict__ bias1,
    const float* __restrict__ W2, const float* __restrict__ bias2,
    float* __restrict__ etl_out, float* __restrict__ scores_out)
{
    __shared__ __align__(16) float wbufT[2][64 * WPITCH];  // 17.4 KB
    __shared__ float hbuf[128][65];                        // 33.3 KB
    __shared__ float ered[128][2][NTYPES];                 //  7.2 KB

    const int tid   = threadIdx.x;
    const int wave  = tid >> 5;
    const int lane  = tid & 31;
    const int lhalf = lane >> 4;
    const int l15   = lane & 15;
    const int mBlock = blockIdx.x * 128;
    const int mWave  = mBlock + wave * 16;
    const int krow   = lhalf * 2;

    const int myRow   = tid >> 1;
    const int rowHalf = tid & 1;

    const float* Arow = seq + (size_t)(mWave + l15) * Dn;

    float accE[NTYPES];
#pragma unroll
    for (int j = 0; j < NTYPES; ++j) accE[j] = 0.0f;

    for (int nc = 0; nc < Dn / 64; ++nc) {
        const int n0 = nc * 64;
        v8f c0 = {0,0,0,0,0,0,0,0}, c1 = {0,0,0,0,0,0,0,0};
        v8f c2 = {0,0,0,0,0,0,0,0}, c3 = {0,0,0,0,0,0,0,0};

        stage_w_async(W1, Dn, 0, n0, wbufT[0], tid);

        for (int ic = 0; ic < Dn / 32; ++ic) {
            const int k0 = ic * 32;
            wait_async0();
            __syncthreads();
            if (ic + 1 < Dn / 32)
                stage_w_async(W1, Dn, k0 + 32, n0, wbufT[(ic + 1) & 1], tid);

            const float* bb = wbufT[ic & 1];

            v2f areg[8];
#pragma unroll
            for (int s = 0; s < 8; ++s)
                areg[s] = *(const v2f*)(Arow + k0 + s * 4 + krow);

#pragma unroll
            for (int s = 0; s < 8; ++s) {
                const int kk = s * 4;
                v2f b0 = *(const v2f*)(bb + (l15)      * WPITCH + kk + krow);
                v2f b1 = *(const v2f*)(bb + (16 + l15) * WPITCH + kk + krow);
                v2f b2 = *(const v2f*)(bb + (32 + l15) * WPITCH + kk + krow);
                v2f b3 = *(const v2f*)(bb + (48 + l15) * WPITCH + kk + krow);
                c0 = wmma_f32(areg[s], b0, c0);
                c1 = wmma_f32(areg[s], b1, c1);
                c2 = wmma_f32(areg[s], b2, c2);
                c3 = wmma_f32(areg[s], b3, c3);
            }
        }

        __syncthreads();
        // bias + gelu, deposit h tile into LDS
#pragma unroll
        for (int r = 0; r < 8; ++r) {
            const int rowt = wave * 16 + r + 8 * lhalf;
            float x;
            x = c0[r] + bias1[n0 + l15];        hbuf[rowt][l15]      = gelu_exact(x);
            x = c1[r] + bias1[n0 + 16 + l15];   hbuf[rowt][16 + l15] = gelu_exact(x);
            x = c2[r] + bias1[n0 + 32 + l15];   hbuf[rowt][32 + l15] = gelu_exact(x);
            x = c3[r] + bias1[n0 + 48 + l15];   hbuf[rowt][48 + l15] = gelu_exact(x);
        }
        __syncthreads();

        // skinny GEMM: etl[row][j] += h[row][n] * W2[n][j]
        for (int nn = 0; nn < 32; ++nn) {
            const float hv = hbuf[myRow][rowHalf * 32 + nn];
            const int wrow = n0 + rowHalf * 32 + nn;
#pragma unroll
            for (int j = 0; j < NTYPES; ++j)
                accE[j] += hv * W2[(size_t)wrow * NTYPES + j];
        }
        __syncthreads();   // hbuf/wbuf reuse across nc
    }

#pragma unroll
    for (int j = 0; j < NTYPES; ++j) ered[myRow][rowHalf][j] = accE[j];
    __syncthreads();

    if (rowHalf == 0) {
        const int grow = mBlock + myRow;
        float lg[NTYPES];
        float mx = -1e30f;
#pragma unroll
        for (int j = 0; j < NTYPES; ++j) {
            float v = ered[myRow][0][j] + ered[myRow][1][j] + bias2[j];
            lg[j] = v;
            etl_out[(size_t)grow * NTYPES + j] = v;
            mx = fmaxf(mx, v);
        }
        float s = 0.0f;
#pragma unroll
        for (int j = 0; j < NTYPES; ++j) s += expf(lg[j] - mx);
        const float p0 = expf(lg[0] - mx) / s;
        scores_out[grow] = 1.0f - p0;
    }
}

// =====================================================================
// Kernel 2: generic WMMA GEMM  C[M,N] = A[M,Kd] @ W[Kd,N] + bias
// macro-tile 128(M) x 64(N); async double-buffered transposed W tile.
// Requires M % 128 == 0, Kd % 32 == 0, N % 64 == 0 (all uses comply).
// =====================================================================
__global__ __launch_bounds__(256) void wmma_gemm_tiled(
    const float* __restrict__ A, const float* __restrict__ W,
    const float* __restrict__ bias, float* __restrict__ C,
    int M, int N, int Kd)
{
    __shared__ __align__(16) float wbufT[2][64 * WPITCH];

    const int tid   = threadIdx.x;
    const int wave  = tid >> 5;
    const int lane  = tid & 31;
    const int lhalf = lane >> 4;
    const int l15   = lane & 15;
    const int m0 = blockIdx.y * 128 + wave * 16;
    const int n0 = blockIdx.x * 64;
    const int krow = lhalf * 2;

    const float* Arow = A + (size_t)(m0 + l15) * Kd;

    v8f c0 = {0,0,0,0,0,0,0,0}, c1 = {0,0,0,0,0,0,0,0};
    v8f c2 = {0,0,0,0,0,0,0,0}, c3 = {0,0,0,0,0,0,0,0};

    const int nch = Kd / 32;
    stage_w_async(W, N, 0, n0, wbufT[0], tid);

    for (int ic = 0; ic < nch; ++ic) {
        const int k0 = ic * 32;
        wait_async0();
        __syncthreads();
        if (ic + 1 < nch)
            stage_w_async(W, N, k0 + 32, n0, wbufT[(ic + 1) & 1], tid);

        const float* bb = wbufT[ic & 1];

        v2f areg[8];
#pragma unroll
        for (int s = 0; s < 8; ++s)
            areg[s] = *(const v2f*)(Arow + k0 + s * 4 + krow);

#pragma unroll
        for (int s = 0; s < 8; ++s) {
            const int kk = s * 4;
            v2f b0 = *(const v2f*)(bb + (l15)      * WPITCH + kk + krow);
            v2f b1 = *(const v2f*)(bb + (16 + l15) * WPITCH + kk + krow);
            v2f b2 = *(const v2f*)(bb + (32 + l15) * WPITCH + kk + krow);
            v2f b3 = *(const v2f*)(bb + (48 + l15) * WPITCH + kk + krow);
            c0 = wmma_f32(areg[s], b0, c0);
            c1 = wmma_f32(areg[s], b1, c1);
            c2 = wmma_f32(areg[s], b2, c2);
            c3 = wmma_f32(areg[s], b3, c3);
        }
    }

#pragma unroll
    for (int r = 0; r < 8; ++r) {
        const int row = m0 + r + 8 * lhalf;
        int col;
        col = n0 + l15;       C[(size_t)row * N + col] = c0[r] + bias[col];
        col = n0 + 16 + l15;  C[(size_t)row * N + col] = c1[r] + bias[col];
        col = n0 + 32 + l15;  C[(size_t)row * N + col] = c2[r] + bias[col];
        col = n0 + 48 + l15;  C[(size_t)row * N + col] = c3[r] + bias[col];
    }
}

// =====================================================================
// Kernel 3: per-batch event selection (threshold + top-K + sort + fallback)
// =====================================================================
__global__ __launch_bounds__(256) void select_events_kernel(
    const float* __restrict__ scores,
    float* __restrict__ mask_out, float* __restrict__ t_out,
    int* __restrict__ t_int, int* __restrict__ mask_int)
{
    __shared__ float sc[Tn];
    __shared__ float rv[256];
    __shared__ int   ri[256];
    __shared__ int   cnt;
    __shared__ float topv[Kn];
    __shared__ int   topi[Kn];

    const int b = blockIdx.x;
    const int tid = threadIdx.x;
    if (tid == 0) cnt = 0;
    __syncthreads();

    int local = 0;
    for (int i = tid; i < Tn; i += 256) {
        float s = scores[(size_t)b * Tn + i];
        bool above = (s >= THRv);
        local += above ? 1 : 0;
        sc[i] = above ? s : NEGV;
    }
    atomicAdd(&cnt, local);
    __syncthreads();

    for (int k = 0; k < Kn; ++k) {
        float bv = -3.0e30f;
        int bi = Tn;
        for (int i = tid; i < Tn; i += 256) {
            float v = sc[i];
            if (v > bv) { bv = v; bi = i; }     // strided scan: first hit = lowest idx
        }
        rv[tid] = bv; ri[tid] = bi;
        __syncthreads();
        for (int s = 128; s > 0; s >>= 1) {
            if (tid < s) {
                float ov = rv[tid + s]; int oi = ri[tid + s];
                if (ov > rv[tid] || (ov == rv[tid] && oi < ri[tid])) { rv[tid] = ov; ri[tid] = oi; }
            }
            __syncthreads();
        }
        if (tid == 0) {
            topv[k] = rv[0];
            topi[k] = ri[0];
            sc[ri[0]] = -2.0e30f;   // remove from future rounds
        }
        __syncthreads();
    }

    if (tid == 0) {
        int times[Kn];
        for (int k = 0; k < Kn; ++k)
            times[k] = (topv[k] > NEGV * 0.5f) ? topi[k] : Tn;
        for (int i = 1; i < Kn; ++i) {          // insertion sort ascending
            int v = times[i], j = i - 1;
            while (j >= 0 && times[j] > v) { times[j + 1] = times[j]; --j; }
            times[j + 1] = v;
        }
        const int count = cnt;
        const bool fb = (count == 0);
        const int n = fb ? Kn : (count < Kn ? count : Kn);
        for (int j = 0; j < Kn; ++j) {
            int tj = fb ? j : times[j];
            int mk = (j < n) ? 1 : 0;
            int tv = mk ? tj : 0;
            mask_out[b * Kn + j] = (float)mk;
            t_out[b * Kn + j]    = (float)tv;
            mask_int[b * Kn + j] = mk;
            t_int[b * Kn + j]    = tv;
        }
    }
}

// =====================================================================
// Kernel 4: gather selected rows -> raw[256][2112], holder softmax, type argmax
// =====================================================================
__global__ __launch_bounds__(256) void gather_prep_kernel(
    const float* __restrict__ seq, const float* __restrict__ st,
    const float* __restrict__ hl, const float* __restrict__ etl,
    const int* __restrict__ t_int, const int* __restrict__ mask_int,
    float* __restrict__ raw, int* __restrict__ type_int,
    float* __restrict__ typeids_out)
{
    __shared__ float hv[En];
    __shared__ float ssum;

    const int slot = blockIdx.x;          // 0..255
    const int tid = threadIdx.x;
    const int b = slot / Kn;
    const int t = t_int[slot];
    const int mk = mask_int[slot];

    const size_t rowoff = ((size_t)b * Tn + t);
    const float* sp  = seq + rowoff * Dn;
    const float* stp = st  + rowoff * Dn;
    float* rp = raw + (size_t)slot * (2 * Dn + En);

    for (int d = tid; d < Dn; d += 256) {
        rp[d]      = sp[d];
        rp[Dn + d] = stp[d];
    }

    if (tid < En) hv[tid] = hl[rowoff * En + tid];
    __syncthreads();
    if (tid == 0) {
        float mx = hv[0];
        for (int e = 1; e < En; ++e) mx = fmaxf(mx, hv[e]);
        float s = 0.0f;
        for (int e = 0; e < En; ++e) { float ex = expf(hv[e] - mx); hv[e] = ex; s += ex; }
        ssum = s;
        const float* ep = etl + rowoff * NTYPES;
        int bi = 0; float bvv = ep[0];
        for (int j = 1; j < NTYPES; ++j) if (ep[j] > bvv) { bvv = ep[j]; bi = j; }
        type_int[slot] = bi;
        typeids_out[slot] = mk ? (float)bi : 0.0f;
    }
    __syncthreads();
    if (tid < En) rp[2 * Dn + tid] = hv[tid] / ssum;
}

// =====================================================================
// Kernel 5: entries epilogue: add type/time embeddings, apply mask
// =====================================================================
__global__ __launch_bounds__(256) void entry_finish_kernel(
    const float* __restrict__ gemmOut,
    const int* __restrict__ t_int, const int* __restrict__ mask_int,
    const int* __restrict__ type_int,
    const float* __restrict__ type_embed, const float* __restrict__ time_embed,
    float* __restrict__ entries)
{
    const int idx = blockIdx.x * 256 + threadIdx.x;   // 0..262143
    const int d = idx & (Dn - 1);
    const int slot = idx >> 10;
    if (mask_int[slot]) {
        const int ty = type_int[slot];
        int tt = t_int[slot]; if (tt > MAXTIME - 1) tt = MAXTIME - 1;
        entries[idx] = gemmOut[idx]
                     + type_embed[(size_t)ty * Dn + d]
                     + time_embed[(size_t)tt * Dn + d];
    } else {
        entries[idx] = 0.0f;
    }
}

// =====================================================================
extern "C" void kernel_launch(void* const* d_in, const int* in_sizes, int n_in,
                              void* d_out, int out_size, void* d_ws, size_t ws_size,
                              hipStream_t stream)
{
    const float* seq       = (const float*)d_in[0];
    const float* st        = (const float*)d_in[1];
    const float* hl        = (const float*)d_in[2];
    const float* th_w1     = (const float*)d_in[3];
    const float* th_b1     = (const float*)d_in[4];
    const float* th_w2     = (const float*)d_in[5];
    const float* th_b2     = (const float*)d_in[6];
    const float* prev_w    = (const float*)d_in[7];
    const float* prev_b    = (const float*)d_in[8];
    const float* next_w    = (const float*)d_in[9];
    const float* next_b    = (const float*)d_in[10];
    const float* type_emb  = (const float*)d_in[11];
    const float* time_emb  = (const float*)d_in[12];
    const float* entry_w   = (const float*)d_in[13];
    const float* entry_b   = (const float*)d_in[14];

    // ---- d_out layout (tuple concatenated flat, all f32) ----
    float* out = (float*)d_out;
    float* o_entries = out;                                   // 8*32*1024
    float* o_mask    = o_entries + (size_t)Bn * Kn * Dn;      // 256
    float* o_t       = o_mask + Bn * Kn;                      // 256
    float* o_tid     = o_t + Bn * Kn;                         // 256
    float* o_etl     = o_tid + Bn * Kn;                       // 8*4096*7
    float* o_prev    = o_etl + (size_t)BT * NTYPES;           // 8*4096*64
    float* o_next    = o_prev + (size_t)BT * En;
    float* o_scores  = o_next + (size_t)BT * En;              // 8*4096

    // ---- workspace layout (~3.2 MB) ----
    char* w = (char*)d_ws;
    float* ws_raw = (float*)w;  w += (size_t)(Bn * Kn) * (2 * Dn + En) * sizeof(float);
    float* ws_eg  = (float*)w;  w += (size_t)(Bn * Kn) * Dn * sizeof(float);
    int* ws_tint  = (int*)w;    w += Bn * Kn * sizeof(int);
    int* ws_mint  = (int*)w;    w += Bn * Kn * sizeof(int);
    int* ws_type  = (int*)w;    w += Bn * Kn * sizeof(int);

    // 1) fused h -> event_type_logits -> event_scores
    fused_h_etl_kernel<<<BT / 128, 256, 0, stream>>>(
        seq, th_w1, th_b1, th_w2, th_b2, o_etl, o_scores);

    // 2/3) prev / next holder logits
    wmma_gemm_tiled<<<dim3(1, BT / 128), 256, 0, stream>>>(
        seq, prev_w, prev_b, o_prev, BT, En, Dn);
    wmma_gemm_tiled<<<dim3(1, BT / 128), 256, 0, stream>>>(
        seq, next_w, next_b, o_next, BT, En, Dn);

    // 4) per-batch selection
    select_events_kernel<<<Bn, 256, 0, stream>>>(
        o_scores, o_mask, o_t, ws_tint, ws_mint);

    // 5) gather + holder softmax + type argmax
    gather_prep_kernel<<<Bn * Kn, 256, 0, stream>>>(
        seq, st, hl, o_etl, ws_tint, ws_mint, ws_raw, ws_type, o_tid);

    // 6) entry projection: raw[256,2112] @ entry_w[2112,1024] + entry_b
    wmma_gemm_tiled<<<dim3(Dn / 64, (Bn * Kn) / 128), 256, 0, stream>>>(
        ws_raw, entry_w, entry_b, ws_eg, Bn * Kn, Dn, 2 * Dn + En);

    // 7) embeddings + mask epilogue
    entry_finish_kernel<<<(Bn * Kn * Dn) / 256, 256, 0, stream>>>(
        ws_eg, ws_tint, ws_mint, ws_type, type_emb, time_emb, o_entries);
}